// RadarEncNO_35450660061782
// MI455X (gfx1250) — compile-verified
//
#include <hip/hip_runtime.h>
#include <hip/hip_bf16.h>

typedef __attribute__((ext_vector_type(2))) float v2f;
typedef __attribute__((ext_vector_type(8))) float v8f;

#define NPTS 43750
#define NR 175
#define KPTS 250
#define CAP1 350000
#define CAP2 69000

// ---------------------------------------------------------------- attention
__global__ __launch_bounds__(256) void attn_kernel(
    const float* __restrict__ vf, const int* __restrict__ coors,
    const float* __restrict__ ele_t, const float* __restrict__ azi_t,
    const float* __restrict__ wq, const float* __restrict__ bq,
    const float* __restrict__ wk, const float* __restrict__ bk,
    const float* __restrict__ wv, const float* __restrict__ bv,
    float* __restrict__ f_att)
{
    __shared__ float s_xin[KPTS * 8];
    __shared__ float s_k[KPTS * 32];
    __shared__ float s_v[KPTS];
    __shared__ float s_wq[224], s_wk[224], s_bq[32], s_bk[32], s_wv[8];

    const int tid = threadIdx.x;
    const int r = blockIdx.x;
    const int i = tid;
    const bool act = (i < KPTS);
    const int gi = r * KPTS + i;

    if (act) {
        float pw = log10f(vf[2 * gi + 0]);   // voxel_features[:, -2] == col 0
        int c1 = coors[4 * gi + 1];
        int c3 = coors[4 * gi + 3];
        s_xin[i * 8 + 0] = pw;
        for (int t = 0; t < 3; ++t) {
            s_xin[i * 8 + 1 + t] = ele_t[c1 * 3 + t];
            s_xin[i * 8 + 4 + t] = azi_t[c3 * 3 + t];
        }
    }

    const float scale = 0.17677669529663687f;  // 32^-0.5
    float x = 0.0f;

    for (int l = 0; l < 4; ++l) {
        __syncthreads();
        for (int t = tid; t < 224; t += 256) {
            s_wq[t] = wq[l * 224 + t];
            s_wk[t] = wk[l * 224 + t];
        }
        if (tid < 32) { s_bq[tid] = bq[l * 32 + tid]; s_bk[tid] = bk[l * 32 + tid]; }
        if (tid < 7)  s_wv[tid] = wv[l * 7 + tid];
        if (tid == 7) s_wv[7] = bv[l];
        __syncthreads();

        float q[32];
        if (act) {
            float xin[7];
            for (int d = 0; d < 7; ++d) xin[d] = s_xin[i * 8 + d];
            for (int o = 0; o < 32; ++o) {
                float aq = s_bq[o], ak = s_bk[o];
                for (int d = 0; d < 7; ++d) {
                    aq += xin[d] * s_wq[d * 32 + o];
                    ak += xin[d] * s_wk[d * 32 + o];
                }
                q[o] = aq;
                s_k[i * 32 + o] = ak;
            }
            float vv = s_wv[7];
            for (int d = 0; d < 7; ++d) vv += xin[d] * s_wv[d];
            s_v[i] = vv;
        }
        __syncthreads();

        if (act) {
            float mx = -1e30f, den = 0.0f, num = 0.0f;
            for (int j = 0; j < KPTS; ++j) {
                float s = 0.0f;
                for (int o = 0; o < 32; ++o) s += q[o] * s_k[j * 32 + o];
                s *= scale;
                float nm = fmaxf(mx, s);
                float c = expf(mx - nm);
                float e = expf(s - nm);
                den = den * c + e;
                num = num * c + e * s_v[j];
                mx = nm;
            }
            x = num / den;
        }
        __syncthreads();
        if (act) s_xin[i * 8 + 0] = x;
    }
    if (act) f_att[gi] = x;
}

// -------------------------------------------------- mark downsample outputs
__device__ inline void mark_outputs(int p0, int p1, int p2,
                                    unsigned char* om, int D0, int D1, int D2)
{
    #pragma unroll
    for (int d0 = 0; d0 < 3; ++d0)
    #pragma unroll
    for (int d1 = 0; d1 < 3; ++d1)
    #pragma unroll
    for (int d2 = 0; d2 < 3; ++d2) {
        int t0 = p0 + 1 - d0, t1 = p1 + 1 - d1, t2 = p2 + 1 - d2;
        if ((t0 & 1) || (t1 & 1) || (t2 & 1)) continue;  // (also filters t==-1)
        int o0 = t0 >> 1, o1 = t1 >> 1, o2 = t2 >> 1;
        if (o0 < D0 && o1 < D1 && o2 < D2)
            om[(o0 * D1 + o1) * D2 + o2] = 1;
    }
}

// ----------------------------------------- input MLP + groupnorm + lut0/om1
__global__ __launch_bounds__(256) void f0_kernel(
    const int* __restrict__ coors, const float* __restrict__ f_att,
    const float* __restrict__ w_in, const float* __restrict__ b_in,
    const float* __restrict__ gg, const float* __restrict__ gb,
    float* __restrict__ f0, int* __restrict__ lut0, unsigned char* __restrict__ om1)
{
    int i = blockIdx.x * 256 + threadIdx.x;
    if (i >= NPTS) return;
    int c1 = coors[4 * i + 1], c2 = coors[4 * i + 2], c3 = coors[4 * i + 3];
    int ele = 2 * c1 - 9, rng = 2 * c2, azi = 2 * c3 + 149;
    float f[4] = {(float)ele, (float)rng, (float)azi, f_att[i]};
    float y[16];
    for (int o = 0; o < 16; ++o) {
        float a = b_in[o];
        for (int d = 0; d < 4; ++d) a += f[d] * w_in[d * 16 + o];
        y[o] = a;
    }
    for (int g = 0; g < 8; ++g) {
        float a0 = y[2 * g], a1 = y[2 * g + 1];
        float mu = 0.5f * (a0 + a1);
        float e0 = a0 - mu, e1 = a1 - mu;
        float var = 0.5f * (e0 * e0 + e1 * e1);
        float rs = rsqrtf(var + 1e-5f);
        for (int t = 0; t < 2; ++t) {
            int ch = 2 * g + t;
            float v = (y[ch] - mu) * rs * gg[ch] + gb[ch];
            f0[(size_t)i * 16 + ch] = fmaxf(v, 0.0f);
        }
    }
    int flat = (ele * 350 + rng) * 512 + azi;
    atomicMax(&lut0[flat], i);
    mark_outputs(ele, rng, azi, om1, 28, 175, 256);
}

__global__ __launch_bounds__(256) void mark2_kernel(
    const int* __restrict__ oc1, const int* __restrict__ cnt,
    unsigned char* __restrict__ om2)
{
    int row = blockIdx.x * 256 + threadIdx.x;
    if (row >= *cnt) return;
    mark_outputs(oc1[3 * row], oc1[3 * row + 1], oc1[3 * row + 2], om2, 14, 88, 128);
}

// ------------------------------------- deterministic row-major compaction
__global__ __launch_bounds__(1024) void compact_kernel(
    const unsigned char* __restrict__ om, int n, int d1, int d2, int cap,
    int* __restrict__ oc, int* __restrict__ lut, int* __restrict__ cnt)
{
    __shared__ int s[1024];
    __shared__ int carry;
    int tid = threadIdx.x;
    if (tid == 0) carry = 0;
    __syncthreads();
    for (int base = 0; base < n; base += 1024) {
        int pos = base + tid;
        int f = (pos < n) ? (int)om[pos] : 0;
        s[tid] = f;
        __syncthreads();
        for (int off = 1; off < 1024; off <<= 1) {
            int t = (tid >= off) ? s[tid - off] : 0;
            __syncthreads();
            s[tid] += t;
            __syncthreads();
        }
        int excl = s[tid] - f;
        int total = s[1023];
        int row = carry + excl;
        if (f && row < cap) {
            lut[pos] = row;
            int i2 = pos % d2, i1 = (pos / d2) % d1, i0 = pos / (d1 * d2);
            oc[3 * row] = i0; oc[3 * row + 1] = i1; oc[3 * row + 2] = i2;
        }
        __syncthreads();
        if (tid == 0) carry += total;
        __syncthreads();
    }
    if (tid == 0) *cnt = carry < cap ? carry : cap;
}

// -------------------------------- weight repack into per-lane WMMA layout
// src: (27, CIN, COUT)  ->  dst v2f[((k*KC4 + kc4)*NT + nt)*32 + lane]
//   = { w[k][kc4*4+2h][nt*16+m], w[k][kc4*4+2h+1][nt*16+m] },  h=lane>>4, m=lane&15
__global__ __launch_bounds__(256) void repack_kernel(
    const float* __restrict__ w, float* __restrict__ wp, int CIN, int COUT, int total)
{
    int i = blockIdx.x * 256 + threadIdx.x;
    if (i >= total) return;
    int lane = i & 31;
    int tmp = i >> 5;
    int NT = COUT >> 4;
    int nt = tmp % NT; tmp /= NT;
    int KC4 = CIN >> 2;
    int kc4 = tmp % KC4;
    int k = tmp / KC4;
    int h = lane >> 4, m = lane & 15;
    int r0 = kc4 * 4 + 2 * h;
    int n = nt * 16 + m;
    wp[2 * i]     = w[((size_t)k * CIN + r0) * COUT + n];
    wp[2 * i + 1] = w[((size_t)k * CIN + r0 + 1) * COUT + n];
}

// ------------------------------------------ sparse conv GEMM (WMMA f32)
// one wave per 16-row tile; 8 waves per block; packed weights.
// mode 0: bn+relu    mode 1: bn + residual + relu    mode 2: +bias (raw)
template <int CIN, int COUT, int STRIDE>
__global__ __launch_bounds__(256) void conv_kernel(
    const float* __restrict__ fin, const float* __restrict__ wp,
    const int* __restrict__ oc, const int* __restrict__ countPtr,
    const int* __restrict__ lut, int d0, int d1, int d2,
    float* __restrict__ fout, const float* __restrict__ bnp,
    const float* __restrict__ resid, const float* __restrict__ bias, int mode)
{
    const int count = countPtr[0];
    const int wave = threadIdx.x >> 5;
    const int lane = threadIdx.x & 31;
    const int tile = blockIdx.x * 8 + wave;
    const int rowBase = tile * 16;
    if (rowBase >= count) return;

    const int h = lane >> 4;
    const int m = lane & 15;
    const int row = rowBase + m;
    const bool rv = (row < count);
    const int cr = rv ? row : (count - 1);          // clamped, always-valid load
    const int e = oc[3 * cr], r = oc[3 * cr + 1], a = oc[3 * cr + 2];
    const int eb = STRIDE * e - 1, rb = STRIDE * r - 1, ab = STRIDE * a - 1;

    constexpr int NT = COUT / 16;
    constexpr int KC4 = CIN / 4;
    const v2f* __restrict__ wp2 = (const v2f*)wp;

    v8f acc[NT];
    #pragma unroll
    for (int nt = 0; nt < NT; ++nt)
        #pragma unroll
        for (int j = 0; j < 8; ++j) acc[nt][j] = 0.0f;

    int kk = 0;
    for (int da = 0; da < 3; ++da)
    for (int db = 0; db < 3; ++db)
    for (int dc = 0; dc < 3; ++dc, ++kk) {
        const int n0 = eb + da, n1 = rb + db, n2 = ab + dc;
        const bool inb = rv & ((unsigned)n0 < (unsigned)d0)
                            & ((unsigned)n1 < (unsigned)d1)
                            & ((unsigned)n2 < (unsigned)d2);
        const int c0 = min(max(n0, 0), d0 - 1);
        const int c1 = min(max(n1, 0), d1 - 1);
        const int c2 = min(max(n2, 0), d2 - 1);
        int idx = lut[((size_t)c0 * d1 + c1) * d2 + c2];   // branchless lookup
        if (!inb) idx = -1;
        const bool ok = idx >= 0;
        const v2f* __restrict__ arow =
            (const v2f*)(fin + (size_t)max(idx, 0) * CIN);

        for (int kc4 = 0; kc4 < KC4; ++kc4) {
            v2f t = arow[kc4 * 2 + h];                      // one b64 load
            v2f av;
            av.x = ok ? t.x : 0.0f;
            av.y = ok ? t.y : 0.0f;
            const v2f* __restrict__ wpp =
                wp2 + ((size_t)(kk * KC4 + kc4) * NT) * 32 + lane;
            #pragma unroll
            for (int nt = 0; nt < NT; ++nt) {
                v2f bv = wpp[nt * 32];                     // coalesced b64 load
                acc[nt] = __builtin_amdgcn_wmma_f32_16x16x4_f32(
                    false, av, false, bv, (short)0, acc[nt], false, false);
            }
        }
    }

    #pragma unroll
    for (int nt = 0; nt < NT; ++nt) {
        int ch = nt * 16 + m;
        float g = 1.0f, b = 0.0f, mu = 0.0f, var = 1.0f;
        if (mode < 2) {
            g = bnp[ch]; b = bnp[COUT + ch];
            mu = bnp[2 * COUT + ch]; var = bnp[3 * COUT + ch];
        }
        float rs = rsqrtf(var + 1e-3f);
        #pragma unroll
        for (int v = 0; v < 8; ++v) {
            int orow = rowBase + v + 8 * h;   // D: M = v + 8h, N = m
            if (orow < count) {
                float x = acc[nt][v];
                float y;
                if (mode == 2) {
                    y = x + bias[ch];
                } else {
                    y = (x - mu) * rs * g + b;
                    if (mode == 1) y += resid[(size_t)orow * COUT + ch];
                    y = fmaxf(y, 0.0f);
                }
                fout[(size_t)orow * COUT + ch] = y;
            }
        }
    }
}

// ---------------------------------------- final groupnorm + relu + scatter
__global__ __launch_bounds__(256) void gn_scatter_kernel(
    const float* __restrict__ fo, const int* __restrict__ oc,
    const int* __restrict__ cnt, const float* __restrict__ gg,
    const float* __restrict__ gb, float* __restrict__ out)
{
    int row = blockIdx.x * 256 + threadIdx.x;
    if (row >= *cnt) return;
    int a0 = oc[3 * row], a1 = oc[3 * row + 1], a2 = oc[3 * row + 2];
    const float* x = fo + (size_t)row * 128;
    for (int g = 0; g < 16; ++g) {
        float v[8]; float s = 0.0f;
        for (int t = 0; t < 8; ++t) { v[t] = x[8 * g + t]; s += v[t]; }
        float mu = s * 0.125f, var = 0.0f;
        for (int t = 0; t < 8; ++t) { float d = v[t] - mu; var += d * d; }
        var *= 0.125f;
        float rs = rsqrtf(var + 1e-5f);
        for (int t = 0; t < 8; ++t) {
            int ch = 8 * g + t;
            float y = fmaxf((v[t] - mu) * rs * gg[ch] + gb[ch], 0.0f);
            // out shape (1, 128ch, 128, 88, 14)
            out[(((size_t)ch * 128 + a2) * 88 + a1) * 14 + a0] = y;
        }
    }
}

// ---------------------------------------------------------------- launcher
extern "C" void kernel_launch(void* const* d_in, const int* in_sizes, int n_in,
                              void* d_out, int out_size, void* d_ws, size_t ws_size,
                              hipStream_t stream) {
    (void)in_sizes; (void)n_in; (void)ws_size;
    const float* vf     = (const float*)d_in[0];
    const int*   coors  = (const int*)d_in[1];
    const float* ele_t  = (const float*)d_in[3];
    const float* azi_t  = (const float*)d_in[4];
    const float* wq     = (const float*)d_in[5];
    const float* bq     = (const float*)d_in[6];
    const float* wk     = (const float*)d_in[7];
    const float* bk     = (const float*)d_in[8];
    const float* wv     = (const float*)d_in[9];
    const float* bv     = (const float*)d_in[10];
    const float* w_in   = (const float*)d_in[11];
    const float* b_in   = (const float*)d_in[12];
    const float* gin_g  = (const float*)d_in[13];
    const float* gin_b  = (const float*)d_in[14];
    const float* w_d1   = (const float*)d_in[15];
    const float* bn_d1  = (const float*)d_in[16];
    const float* w_r1   = (const float*)d_in[17];
    const float* bn_r1  = (const float*)d_in[18];
    const float* w_d2   = (const float*)d_in[19];
    const float* bn_d2  = (const float*)d_in[20];
    const float* w_r2   = (const float*)d_in[21];
    const float* bn_r2  = (const float*)d_in[22];
    const float* w_out  = (const float*)d_in[23];
    const float* b_out  = (const float*)d_in[24];
    const float* gout_g = (const float*)d_in[25];
    const float* gout_b = (const float*)d_in[26];
    float* out = (float*)d_out;

    char* p = (char*)d_ws;
    auto alloc = [&](size_t bytes) -> void* {
        void* q = (void*)p;
        p += (bytes + 255) & ~(size_t)255;
        return q;
    };
    const size_t L0 = (size_t)56 * 350 * 512;   // 10,035,200
    const size_t L1 = (size_t)28 * 175 * 256;   //  1,254,400
    const size_t L2 = (size_t)14 * 88 * 128;    //    157,696
    int*  lut0 = (int*)alloc(L0 * 4);
    int*  lut1 = (int*)alloc(L1 * 4);
    int*  lut2 = (int*)alloc(L2 * 4);
    unsigned char* om1 = (unsigned char*)alloc(L1);
    unsigned char* om2 = (unsigned char*)alloc(L2);
    int*  oc1  = (int*)alloc((size_t)CAP1 * 3 * 4);
    int*  oc2  = (int*)alloc((size_t)CAP2 * 3 * 4);
    int*  cnt1 = (int*)alloc(4);
    int*  cnt2 = (int*)alloc(4);
    float* f_att = (float*)alloc((size_t)NPTS * 4);
    float* f0    = (float*)alloc((size_t)NPTS * 16 * 4);
    float* f1a   = (float*)alloc((size_t)CAP1 * 32 * 4);
    float* f1b   = (float*)alloc((size_t)CAP1 * 32 * 4);
    float* f2a   = (float*)alloc((size_t)CAP2 * 128 * 4);
    float* f2b   = (float*)alloc((size_t)CAP2 * 128 * 4);
    // packed weights
    const int T_D1 = 27 * 4 * 2 * 32;       //  6912 v2f
    const int T_R1 = 27 * 8 * 2 * 32;       // 13824 v2f per conv
    const int T_D2 = 27 * 8 * 8 * 32;       // 55296 v2f
    const int T_R2 = 27 * 32 * 8 * 32;      // 221184 v2f per conv
    float* wp_d1  = (float*)alloc((size_t)T_D1 * 8);
    float* wp_r1  = (float*)alloc((size_t)T_R1 * 8 * 4);
    float* wp_d2  = (float*)alloc((size_t)T_D2 * 8);
    float* wp_r2  = (float*)alloc((size_t)T_R2 * 8 * 4);
    float* wp_out = (float*)alloc((size_t)T_R2 * 8);

    hipMemsetAsync(lut0, 0xFF, L0 * 4, stream);
    hipMemsetAsync(lut1, 0xFF, L1 * 4, stream);
    hipMemsetAsync(lut2, 0xFF, L2 * 4, stream);
    hipMemsetAsync(om1, 0, L1, stream);
    hipMemsetAsync(om2, 0, L2, stream);
    hipMemsetAsync(cnt1, 0, 4, stream);
    hipMemsetAsync(cnt2, 0, 4, stream);
    hipMemsetAsync(out, 0, (size_t)out_size * sizeof(float), stream);

    // weight repack (WMMA per-lane layout)
    repack_kernel<<<(T_D1 + 255) / 256, 256, 0, stream>>>(w_d1, wp_d1, 16, 32, T_D1);
    for (int j = 0; j < 4; ++j)
        repack_kernel<<<(T_R1 + 255) / 256, 256, 0, stream>>>(
            w_r1 + (size_t)j * 27 * 32 * 32, wp_r1 + (size_t)j * T_R1 * 2, 32, 32, T_R1);
    repack_kernel<<<(T_D2 + 255) / 256, 256, 0, stream>>>(w_d2, wp_d2, 32, 128, T_D2);
    for (int j = 0; j < 4; ++j)
        repack_kernel<<<(T_R2 + 255) / 256, 256, 0, stream>>>(
            w_r2 + (size_t)j * 27 * 128 * 128, wp_r2 + (size_t)j * T_R2 * 2, 128, 128, T_R2);
    repack_kernel<<<(T_R2 + 255) / 256, 256, 0, stream>>>(w_out, wp_out, 128, 128, T_R2);

    // attention + input MLP
    attn_kernel<<<NR, 256, 0, stream>>>(vf, coors, ele_t, azi_t,
                                        wq, bq, wk, bk, wv, bv, f_att);
    f0_kernel<<<(NPTS + 255) / 256, 256, 0, stream>>>(
        coors, f_att, w_in, b_in, gin_g, gin_b, f0, lut0, om1);

    // level 1
    compact_kernel<<<1, 1024, 0, stream>>>(om1, (int)L1, 175, 256, CAP1, oc1, lut1, cnt1);
    const int blocks1 = (((CAP1 + 15) / 16) + 7) / 8;  // 2735
    conv_kernel<16, 32, 2><<<blocks1, 256, 0, stream>>>(
        f0, wp_d1, oc1, cnt1, lut0, 56, 350, 512, f1a, bn_d1, nullptr, nullptr, 0);

    for (int blk = 0; blk < 2; ++blk) {
        conv_kernel<32, 32, 1><<<blocks1, 256, 0, stream>>>(
            f1a, wp_r1 + (size_t)(blk * 2 + 0) * T_R1 * 2, oc1, cnt1, lut1, 28, 175, 256,
            f1b, bn_r1 + (size_t)(blk * 2 + 0) * 4 * 32, nullptr, nullptr, 0);
        conv_kernel<32, 32, 1><<<blocks1, 256, 0, stream>>>(
            f1b, wp_r1 + (size_t)(blk * 2 + 1) * T_R1 * 2, oc1, cnt1, lut1, 28, 175, 256,
            f1a, bn_r1 + (size_t)(blk * 2 + 1) * 4 * 32, f1a, nullptr, 1);
    }

    // level 2
    mark2_kernel<<<(CAP1 + 255) / 256, 256, 0, stream>>>(oc1, cnt1, om2);
    compact_kernel<<<1, 1024, 0, stream>>>(om2, (int)L2, 88, 128, CAP2, oc2, lut2, cnt2);
    const int blocks2 = (((CAP2 + 15) / 16) + 7) / 8;  // 540
    conv_kernel<32, 128, 2><<<blocks2, 256, 0, stream>>>(
        f1a, wp_d2, oc2, cnt2, lut1, 28, 175, 256, f2a, bn_d2, nullptr, nullptr, 0);

    for (int blk = 0; blk < 2; ++blk) {
        conv_kernel<128, 128, 1><<<blocks2, 256, 0, stream>>>(
            f2a, wp_r2 + (size_t)(blk * 2 + 0) * T_R2 * 2, oc2, cnt2, lut2, 14, 88, 128,
            f2b, bn_r2 + (size_t)(blk * 2 + 0) * 4 * 128, nullptr, nullptr, 0);
        conv_kernel<128, 128, 1><<<blocks2, 256, 0, stream>>>(
            f2b, wp_r2 + (size_t)(blk * 2 + 1) * T_R2 * 2, oc2, cnt2, lut2, 14, 88, 128,
            f2a, bn_r2 + (size_t)(blk * 2 + 1) * 4 * 128, f2a, nullptr, 1);
    }

    // final conv (+bias), then groupnorm + relu + transpose-scatter
    conv_kernel<128, 128, 1><<<blocks2, 256, 0, stream>>>(
        f2a, wp_out, oc2, cnt2, lut2, 14, 88, 128, f2b, nullptr, nullptr, b_out, 2);
    gn_scatter_kernel<<<(CAP2 + 255) / 256, 256, 0, stream>>>(
        f2b, oc2, cnt2, gout_g, gout_b, out);
}